// _IPEXConcatLinearCPU_57690000720334
// MI455X (gfx1250) — compile-verified
//
#include <hip/hip_runtime.h>

typedef __attribute__((ext_vector_type(2))) float v2f;
typedef __attribute__((ext_vector_type(4))) float v4f;
typedef __attribute__((ext_vector_type(8))) float v8f;

// Matches the builtin's declared parameter: int __attribute__((vector_size(16))) *
typedef int async_b128_t __attribute__((vector_size(16)));

// Problem dims (fixed by the harness reference)
constexpr int Mdim = 8192;    // B*S = 4*2048
constexpr int Ndim = 12288;   // NUM_CONCAT * H
constexpr int Kdim = 4096;    // D
constexpr int Hdim = 4096;    // per-output hidden

// Tiling
constexpr int MT  = 128;          // block M tile
constexpr int NT  = 128;          // block N tile
constexpr int KT  = 16;           // K step (double-buffered)
constexpr int LDT = 20;           // padded row stride (floats): 80B rows -> 16B aligned, bank-conflict-free
constexpr int NTILES = Kdim / KT; // 256

#if __has_builtin(__builtin_amdgcn_global_load_async_to_lds_b128)
#define USE_ASYNC_LDS 1
#else
#define USE_ASYNC_LDS 0
#endif

#if USE_ASYNC_LDS
template <int N>
__device__ static inline void wait_asynccnt() {
#if __has_builtin(__builtin_amdgcn_s_wait_asynccnt)
    __builtin_amdgcn_s_wait_asynccnt(N);
#else
    asm volatile("s_wait_asynccnt %0" :: "i"(N) : "memory");
#endif
}
#endif // USE_ASYNC_LDS

__global__ __launch_bounds__(256)
void concat_linear_wmma_f32(const float* __restrict__ X,
                            const float* __restrict__ W,
                            const float* __restrict__ Bias,
                            float* __restrict__ Out)
{
    __shared__ float aT[2][MT * LDT];   // X tiles: 128 rows x 16 K, double buffered
    __shared__ float bT[2][NT * LDT];   // W tiles: 128 rows x 16 K, double buffered

    const int tid   = threadIdx.x;
    const int lane  = tid & 31;          // wave32
    const int wave  = tid >> 5;          // 0..7
    const int waveM = wave >> 2;         // 0..1  -> 64-row strip
    const int waveN = wave & 3;          // 0..3  -> 32-col strip

    const int mBlock = blockIdx.y * MT;
    const int nBlock = blockIdx.x * NT;

    const int lhalf = lane >> 4;         // 0/1: selects K pair (WMMA f32 layout)
    const int lmod  = lane & 15;         // row within 16x16 fragment

    v8f acc[4][2];
    const v8f vzero = {0.f,0.f,0.f,0.f,0.f,0.f,0.f,0.f};
#pragma unroll
    for (int i = 0; i < 4; ++i)
#pragma unroll
        for (int j = 0; j < 2; ++j) acc[i][j] = vzero;

    // Cooperative tile-load mapping: thread t -> row t/2, 8 floats at col (t%2)*8
    const int ldRow = tid >> 1;          // 0..127
    const int ldCol = (tid & 1) * 8;     // 0 or 8

    const float* xSrc = X + (size_t)(mBlock + ldRow) * Kdim + ldCol;
    const float* wSrc = W + (size_t)(nBlock + ldRow) * Kdim + ldCol;
    const int ldsOff = ldRow * LDT + ldCol;

    // Compute one K-panel out of LDS buffer `bf`
    auto compute_tile = [&](int bf) {
        const float* aB = aT[bf];
        const float* bB = bT[bf];
#pragma unroll
        for (int kk = 0; kk < KT; kk += 4) {
            const int kf = kk + 2 * lhalf;   // this lane-half's K pair
            v2f afrag[4];
            v2f bfrag[2];
#pragma unroll
            for (int i = 0; i < 4; ++i) {
                const int row = waveM * 64 + i * 16 + lmod;
                afrag[i] = *(const v2f*)&aB[row * LDT + kf];
            }
#pragma unroll
            for (int j = 0; j < 2; ++j) {
                const int row = waveN * 32 + j * 16 + lmod;
                bfrag[j] = *(const v2f*)&bB[row * LDT + kf];
            }
#pragma unroll
            for (int i = 0; i < 4; ++i)
#pragma unroll
                for (int j = 0; j < 2; ++j)
                    acc[i][j] = __builtin_amdgcn_wmma_f32_16x16x4_f32(
                        false, afrag[i], false, bfrag[j],
                        (short)0, acc[i][j], false, false);
        }
    };

#if USE_ASYNC_LDS
    // Async double-buffered pipeline: HBM -> LDS via ASYNCcnt path, no VGPR staging.
    auto issue_tile = [&](int bf, int kb) {
        // 4 async b128 loads per thread (wave issues 4 instructions -> ASYNCcnt += 4)
        __builtin_amdgcn_global_load_async_to_lds_b128(
            (async_b128_t*)(xSrc + kb),     (async_b128_t*)&aT[bf][ldsOff],     0, 0);
        __builtin_amdgcn_global_load_async_to_lds_b128(
            (async_b128_t*)(xSrc + kb + 4), (async_b128_t*)&aT[bf][ldsOff + 4], 0, 0);
        __builtin_amdgcn_global_load_async_to_lds_b128(
            (async_b128_t*)(wSrc + kb),     (async_b128_t*)&bT[bf][ldsOff],     0, 0);
        __builtin_amdgcn_global_load_async_to_lds_b128(
            (async_b128_t*)(wSrc + kb + 4), (async_b128_t*)&bT[bf][ldsOff + 4], 0, 0);
    };

    issue_tile(0, 0);
    for (int t = 0; t < NTILES; ++t) {
        const int cur = t & 1;
        if (t + 1 < NTILES) {
            issue_tile(1 - cur, (t + 1) * KT);  // prefetch next panel into other buffer
            wait_asynccnt<4>();                 // this wave's tile-t loads (4) complete
        } else {
            wait_asynccnt<0>();
        }
        __syncthreads();                        // all waves' loads for tile t landed
        compute_tile(cur);
        __syncthreads();                        // safe to overwrite buf `cur` next round
    }
#else
    // Fallback: synchronous staging through VGPRs.
    for (int t = 0; t < NTILES; ++t) {
        const int kb = t * KT;
        v4f xa0 = *(const v4f*)(xSrc + kb);
        v4f xa1 = *(const v4f*)(xSrc + kb + 4);
        v4f wb0 = *(const v4f*)(wSrc + kb);
        v4f wb1 = *(const v4f*)(wSrc + kb + 4);
        __syncthreads();   // previous tile's readers done
        *(v4f*)&aT[0][ldsOff]     = xa0;
        *(v4f*)&aT[0][ldsOff + 4] = xa1;
        *(v4f*)&bT[0][ldsOff]     = wb0;
        *(v4f*)&bT[0][ldsOff + 4] = wb1;
        __syncthreads();   // tile visible to all waves
        compute_tile(0);
    }
#endif

    // Epilogue: bias add + split-chunk addressing.
    // C/D layout: VGPR r, lanes 0-15 -> (M=r, N=lane); lanes 16-31 -> (M=8+r, N=lane-16)
    const int mW = mBlock + waveM * 64;
    const int nW = nBlock + waveN * 32;
#pragma unroll
    for (int j = 0; j < 2; ++j) {
        const int n     = nW + j * 16 + lmod;
        const int chunk = n / Hdim;      // which of the 3 outputs
        const int col   = n % Hdim;
        const float bias = Bias[n];
        float* outBase = Out + (size_t)chunk * ((size_t)Mdim * Hdim) + col;
#pragma unroll
        for (int i = 0; i < 4; ++i) {
            const int mBase = mW + i * 16 + 8 * lhalf;
#pragma unroll
            for (int r = 0; r < 8; ++r) {
                outBase[(size_t)(mBase + r) * Hdim] = acc[i][j][r] + bias;
            }
        }
    }
}

extern "C" void kernel_launch(void* const* d_in, const int* in_sizes, int n_in,
                              void* d_out, int out_size, void* d_ws, size_t ws_size,
                              hipStream_t stream) {
    const float* X    = (const float*)d_in[0];   // [4,2048,4096]
    const float* W    = (const float*)d_in[1];   // [12288,4096]
    const float* Bias = (const float*)d_in[2];   // [12288]
    float* Out        = (float*)d_out;           // 3 x [4,2048,4096] flat

    dim3 grid(Ndim / NT, Mdim / MT);  // (96, 64)
    concat_linear_wmma_f32<<<grid, 256, 0, stream>>>(X, W, Bias, Out);
}